// ScatterLoss_5763846111471
// MI455X (gfx1250) — compile-verified
//
#include <hip/hip_runtime.h>
#include <hip/hip_bf16.h>

typedef __attribute__((ext_vector_type(2))) float v2f;
typedef __attribute__((ext_vector_type(8))) float v8f;

#define N_ROWS 8192
#define DIM    512
#define NCLS   100
#define MROWS  112            // 7 * 16, zero-padded
#define NTILE  7

// Workspace layout
//   float M[MROWS*DIM]                    : class sums (padded rows stay zero)
//   unsigned counts[128]                  : label histogram
//   float invn[N_ROWS]                    : per-row 1/||e_i||
#define WS_M_ELEMS     (MROWS * DIM)
#define WS_CNT_OFF     (WS_M_ELEMS)
#define WS_INVN_OFF    (WS_M_ELEMS + 128)
#define WS_ZERO_ELEMS  (WS_M_ELEMS + 128)

// ---------------------------------------------------------------- zero ws
__global__ void zero_ws_kernel(float* ws) {
    int i = blockIdx.x * blockDim.x + threadIdx.x;
    if (i < WS_ZERO_ELEMS) ws[i] = 0.0f;
}

// ------------------------------------------- row norms + label histogram
// one wave (32 lanes) per row; 1024 blocks x 256 threads (8 waves/block)
__global__ void norm_hist_kernel(const float* __restrict__ e,
                                 const long long* __restrict__ y,
                                 float* __restrict__ invn,
                                 unsigned* __restrict__ counts) {
    int lane = threadIdx.x & 31;
    int wave = threadIdx.x >> 5;
    int row  = blockIdx.x * 8 + wave;
    const float* r = e + (size_t)row * DIM;
    float s = 0.0f;
    #pragma unroll
    for (int j = 0; j < DIM / 32; ++j) {
        float v = r[lane + j * 32];
        s += v * v;
    }
    #pragma unroll
    for (int off = 16; off; off >>= 1) s += __shfl_xor(s, off);
    if (lane == 0) {
        invn[row] = __frsqrt_rn(s);
        atomicAdd(&counts[(int)y[row]], 1u);
    }
}

// ------------------- scatter normalized rows into class-sum matrix M
// grid: (4 column slices of 128) x (16 row groups of 512). 50KB LDS tile.
__global__ void scatter_kernel(const float* __restrict__ e,
                               const long long* __restrict__ y,
                               const float* __restrict__ invn,
                               float* __restrict__ M) {
    __shared__ float sM[NCLS * 128];
    int t = threadIdx.x;              // 256 threads
    for (int i = t; i < NCLS * 128; i += 256) sM[i] = 0.0f;
    __syncthreads();

    int slice   = blockIdx.x;         // 0..3 -> columns [slice*128, +128)
    int rg      = blockIdx.y;         // 0..15 -> rows [rg*512, +512)
    int col     = t & 127;
    int half    = t >> 7;             // two rows in flight per pass
    int rowbase = rg * 512;

    for (int p = 0; p < 256; ++p) {
        int row   = rowbase + p * 2 + half;
        int label = (int)y[row];
        float v   = e[(size_t)row * DIM + slice * 128 + col] * invn[row];
        atomicAdd(&sM[label * 128 + col], v);   // ds_add_f32
    }
    __syncthreads();

    for (int i = t; i < NCLS * 128; i += 256) {
        int c = i >> 7, cc = i & 127;
        atomicAdd(&M[(size_t)c * DIM + slice * 128 + cc], sM[i]);
    }
}

// --------------- WMMA Gram G = M*M^T (112x112, K=512) -> final loss
// single block, 8 waves; 49 tile-pairs strided over waves; f32 16x16x4 WMMA.
__global__ void gram_loss_kernel(const float* __restrict__ M,
                                 const unsigned* __restrict__ counts,
                                 float* __restrict__ out) {
    int lane = threadIdx.x & 31;
    int wave = threadIdx.x >> 5;

    float lsum = 0.0f, ltr = 0.0f;

    for (int tp = wave; tp < NTILE * NTILE; tp += 8) {
        int I = tp / NTILE, J = tp % NTILE;
        // A tile: a[v] = M[I*16 + lane%16][kb + 2*(lane>=16) + v]   (ISA 16x4 layout)
        // B tile: b[v] = M[J*16 + lane%16][kb + 2*(lane>=16) + v]   (symmetric Gram)
        const float* rowA = M + (size_t)(I * 16 + (lane & 15)) * DIM;
        const float* rowB = M + (size_t)(J * 16 + (lane & 15)) * DIM;
        int kofs = (lane >> 4) << 1;          // 0 or 2

        v8f acc = {};
        #pragma unroll 4
        for (int kb = 0; kb < DIM; kb += 4) {
            v2f a, b;
            a.x = rowA[kb + kofs];
            a.y = rowA[kb + kofs + 1];
            b.x = rowB[kb + kofs];
            b.y = rowB[kb + kofs + 1];
            acc = __builtin_amdgcn_wmma_f32_16x16x4_f32(
                false, a, false, b, (short)0, acc, false, false);
        }
        // D layout: lanes 0-15 -> (m=v,      n=lane),
        //           lanes16-31 -> (m=v+8,    n=lane-16)
        float ts = 0.0f;
        #pragma unroll
        for (int v = 0; v < 8; ++v) ts += acc[v];
        lsum += ts;
        if (I == J) {       // uniform per wave
            #pragma unroll
            for (int v = 0; v < 8; ++v)
                ltr += ((lane == v) || (lane == v + 24)) ? acc[v] : 0.0f;
        }
    }

    #pragma unroll
    for (int off = 16; off; off >>= 1) {
        lsum += __shfl_xor(lsum, off);
        ltr  += __shfl_xor(ltr,  off);
    }

    __shared__ float sSum[8], sTr[8];
    if (lane == 0) { sSum[wave] = lsum; sTr[wave] = ltr; }
    __syncthreads();

    if (threadIdx.x == 0) {
        double S = 0.0, Tr = 0.0;
        for (int w = 0; w < 8; ++w) { S += sSum[w]; Tr += sTr[w]; }
        unsigned long long same = 0;
        for (int c = 0; c < NCLS; ++c) {
            unsigned long long n = counts[c];
            same += n * (n - 1ull) / 2ull;
        }
        unsigned long long total =
            (unsigned long long)N_ROWS * (N_ROWS - 1) / 2;
        double cnt  = (double)(total - same);
        double loss = 0.5 * (S - Tr) / cnt;
        out[0] = (float)loss;
    }
}

extern "C" void kernel_launch(void* const* d_in, const int* in_sizes, int n_in,
                              void* d_out, int out_size, void* d_ws, size_t ws_size,
                              hipStream_t stream) {
    const float*     e = (const float*)d_in[0];
    const long long* y = (const long long*)d_in[1];

    float*    ws     = (float*)d_ws;
    float*    M      = ws;
    unsigned* counts = (unsigned*)(ws + WS_CNT_OFF);
    float*    invn   = ws + WS_INVN_OFF;
    float*    out    = (float*)d_out;

    // 1) zero class-sum matrix + histogram
    zero_ws_kernel<<<(WS_ZERO_ELEMS + 255) / 256, 256, 0, stream>>>(ws);
    // 2) row inverse norms + label histogram (one wave per row)
    norm_hist_kernel<<<N_ROWS / 8, 256, 0, stream>>>(e, y, invn, counts);
    // 3) LDS-privatized scatter-accumulate into M (4 col slices x 16 row groups)
    scatter_kernel<<<dim3(4, 16), 256, 0, stream>>>(e, y, invn, M);
    // 4) WMMA Gram of M + final scalar loss
    gram_loss_kernel<<<1, 256, 0, stream>>>(M, counts, out);
}